// SelfModel_76630806495735
// MI455X (gfx1250) — compile-verified
//
#include <hip/hip_runtime.h>
#include <hip/hip_bf16.h>

// ---------------------------------------------------------------------------
// Problem constants (from the reference)
// ---------------------------------------------------------------------------
#define Bb   8
#define Tt   14
#define Gg   (Bb*Tt)        // 112 graphs
#define Nn   512            // nodes
#define Ff   64             // input feature dim
#define HID  128
#define OUT  64
#define K1   410            // ceil(0.8*512)
#define K1P  416            // K1 padded to multiple of 16
#define K2   205            // ceil(0.5*410)
#define Hh   128            // lstm hidden
#define Yy   7
#define INLSTM 65           // OUT+1
#define INPAD  68           // INLSTM padded to multiple of 4

typedef float v2f __attribute__((ext_vector_type(2)));
typedef float v8f __attribute__((ext_vector_type(8)));

// V_WMMA_F32_16X16X4_F32 : D(16x16,f32) = A(16x4,f32) * B(4x16,f32) + C
__device__ __forceinline__ v8f wmma4(v2f a, v2f b, v8f c) {
  return __builtin_amdgcn_wmma_f32_16x16x4_f32(
      /*neg_a=*/false, a, /*neg_b=*/false, b,
      /*c_mod=*/(short)0, c, /*reuse_a=*/false, /*reuse_b=*/false);
}

__device__ __forceinline__ float sigm(float x) { return 1.0f / (1.0f + expf(-x)); }

// ---------------------------------------------------------------------------
// Generic fp32 WMMA GEMM (guard-free): C[M,N] = A[M,K] @ B[K,N] (+bias) (relu)
// Requirements: M % 16 == 0, K % 4 == 0, N % (NT*16) == 0, A/B 8B-aligned rows.
// One wave per block computes a 16 x (NT*16) tile.  grid = (M/16, N/(NT*16))
// ---------------------------------------------------------------------------
template <int NT>
__global__ void gemm_wmma_kernel(const float* __restrict__ A,
                                 const float* __restrict__ Bm,
                                 const float* __restrict__ bias,
                                 float* __restrict__ C,
                                 int K, int lda, int ldb, int ldc, int relu) {
  int m0 = blockIdx.x * 16, n0 = blockIdx.y * (NT * 16);
  int lane = threadIdx.x, half = lane >> 4, l15 = lane & 15;
  int mr = m0 + l15;
  v8f acc[NT];
#pragma unroll
  for (int t = 0; t < NT; ++t) acc[t] = (v8f){};
  const float* arow = A + (size_t)mr * lda + 2 * half;
#pragma unroll 4
  for (int k0 = 0; k0 < K; k0 += 4) {
    v2f a = *(const v2f*)(arow + k0);
    int ka = k0 + 2 * half;
    const float* b0 = Bm + (size_t)ka * ldb + n0 + l15;
#pragma unroll
    for (int t = 0; t < NT; ++t) {
      v2f b;
      b.x = b0[t * 16];
      b.y = b0[t * 16 + ldb];
      acc[t] = wmma4(a, b, acc[t]);
    }
  }
#pragma unroll
  for (int t = 0; t < NT; ++t) {
    int nc = n0 + t * 16 + l15;
    float bb = bias ? bias[nc] : 0.f;
#pragma unroll
    for (int r = 0; r < 8; ++r) {
      int row = m0 + r + 8 * half;
      float v = acc[t][r] + bb;
      if (relu) v = fmaxf(v, 0.f);
      C[(size_t)row * ldc + nc] = v;
    }
  }
}

// ---------------------------------------------------------------------------
// Normalized-adjacency WMMA GEMM (fused GCN aggregate), guard-free inner loop:
//   C[g,i,:] = relu( sum_j dinv[i]*(A[g,i,j]+delta_ij)*dinv[j] * Xw[g,j,:] + b )
// np    : padded node count (multiple of 16; strides & K loop)
// nstore: logical node count (store guard only)
// d     : feature dim, must equal NT*16*gridDim.y
// grid = (np/16, d/(NT*16), G), block = 32
// ---------------------------------------------------------------------------
template <int NT>
__global__ void adj_gemm_kernel(const float* __restrict__ Adj,
                                const float* __restrict__ dinv,
                                const float* __restrict__ Xw,
                                const float* __restrict__ bias,
                                float* __restrict__ C,
                                int np, int nstore, int d) {
  int g  = blockIdx.z;
  int m0 = blockIdx.x * 16, n0 = blockIdx.y * (NT * 16);
  int lane = threadIdx.x, half = lane >> 4, l15 = lane & 15;
  int mr = m0 + l15;     // node i (always < np)
  const float* Ag = Adj + ((size_t)g * np + mr) * np + 2 * half;
  const float* dg = dinv + (size_t)g * np;
  const float* Xg = Xw   + (size_t)g * np * d + n0 + l15;
  float di = dg[mr];
  v8f acc[NT];
#pragma unroll
  for (int t = 0; t < NT; ++t) acc[t] = (v8f){};
#pragma unroll 2
  for (int k0 = 0; k0 < np; k0 += 4) {
    int j0 = k0 + 2 * half;
    v2f ap = *(const v2f*)(Ag + k0);           // A[i][j0], A[i][j0+1]
    v2f dp = *(const v2f*)(dg + j0);           // dinv[j0], dinv[j0+1]
    v2f a;
    a.x = (ap.x + (j0 == mr ? 1.f : 0.f)) * di * dp.x;
    a.y = (ap.y + (j0 + 1 == mr ? 1.f : 0.f)) * di * dp.y;
    const float* b0 = Xg + (size_t)j0 * d;
#pragma unroll
    for (int t = 0; t < NT; ++t) {
      v2f b;
      b.x = b0[t * 16];
      b.y = b0[t * 16 + d];
      acc[t] = wmma4(a, b, acc[t]);
    }
  }
#pragma unroll
  for (int t = 0; t < NT; ++t) {
    int nc = n0 + t * 16 + l15;
    float bb = bias[nc];
#pragma unroll
    for (int r = 0; r < 8; ++r) {
      int row = m0 + r + 8 * half;
      if (row < nstore)
        C[((size_t)g * nstore + row) * d + nc] = fmaxf(acc[t][r] + bb, 0.f);
    }
  }
}

// ---------------------------------------------------------------------------
// ||p1||, ||p2||  (block 0 -> p1 (128), block 1 -> p2 (64))
// ---------------------------------------------------------------------------
__global__ void pnorm_kernel(const float* p1, const float* p2, float* pn) {
  __shared__ float red[128];
  int dim = (blockIdx.x == 0) ? HID : OUT;
  const float* p = (blockIdx.x == 0) ? p1 : p2;
  float v = (threadIdx.x < dim) ? p[threadIdx.x] * p[threadIdx.x] : 0.f;
  red[threadIdx.x] = v;
  __syncthreads();
  for (int s = 64; s > 0; s >>= 1) {
    if ((int)threadIdx.x < s) red[threadIdx.x] += red[threadIdx.x + s];
    __syncthreads();
  }
  if (threadIdx.x == 0) pn[blockIdx.x] = sqrtf(red[0]);
}

// dinv[g,i] = rsqrt( sum_j A[g,i,j] + 1 )     grid = G*n, block = 256
__global__ void degree_kernel(const float* __restrict__ A, float* __restrict__ dinv, int n) {
  __shared__ float red[256];
  long gi = blockIdx.x;                 // g*n + i
  const float* row = A + gi * n;
  float s = 0.f;
  for (int j = threadIdx.x; j < n; j += blockDim.x) s += row[j];
  red[threadIdx.x] = s;
  __syncthreads();
  for (int t = 128; t > 0; t >>= 1) {
    if ((int)threadIdx.x < t) red[threadIdx.x] += red[threadIdx.x + t];
    __syncthreads();
  }
  if (threadIdx.x == 0) dinv[gi] = rsqrtf(red[0] + 1.0f);
}

// score[row] = tanh( dot(Xf[row,:], p) / pn[0] )   one wave32 per row
__global__ void score_kernel(const float* __restrict__ Xf, const float* __restrict__ p,
                             const float* __restrict__ pn, float* __restrict__ sc,
                             int rows, int dim) {
  int warp = blockIdx.x * (blockDim.x >> 5) + (threadIdx.x >> 5);
  int lane = threadIdx.x & 31;
  if (warp >= rows) return;
  const float* row = Xf + (size_t)warp * dim;
  float s = 0.f;
  for (int k = lane; k < dim; k += 32) s += row[k] * p[k];
  for (int off = 16; off > 0; off >>= 1) s += __shfl_xor(s, off);
  if (lane == 0) sc[warp] = tanhf(s / pn[0]);
}

// Per-graph descending bitonic sort of 512 (score,idx) pairs; emit top-k.
// grid = G, block = 256
__global__ void topk_kernel(const float* __restrict__ score,
                            float* __restrict__ vals, int* __restrict__ idx,
                            int n, int k) {
  __shared__ float ss[512];
  __shared__ int   si[512];
  int g = blockIdx.x;
  for (int e = threadIdx.x; e < 512; e += blockDim.x) {
    ss[e] = (e < n) ? score[(size_t)g * n + e] : -3.4e38f;
    si[e] = e;
  }
  __syncthreads();
  for (int kk = 2; kk <= 512; kk <<= 1)
    for (int j = kk >> 1; j > 0; j >>= 1) {
      for (int i = threadIdx.x; i < 512; i += blockDim.x) {
        int l = i ^ j;
        if (l > i) {
          bool desc = ((i & kk) == 0);
          float sa = ss[i], sb = ss[l];
          int ia = si[i], ib = si[l];
          bool keep = (sa > sb) || (sa == sb && ia < ib);   // ties: lower index first
          if (desc ? !keep : keep) { ss[i] = sb; ss[l] = sa; si[i] = ib; si[l] = ia; }
        }
      }
      __syncthreads();
    }
  for (int e = threadIdx.x; e < k; e += blockDim.x) {
    vals[(size_t)g * k + e] = ss[e];
    idx [(size_t)g * k + e] = si[e];
  }
}

// Xp[g,i,:] = H[g, idx[g,i], :] * vals[g,i] for i<k, zeros for k<=i<kp
// grid = (kp, G), block = d
__global__ void gather_x_kernel(const float* __restrict__ Hsrc, const int* __restrict__ idx,
                                const float* __restrict__ vals, float* __restrict__ Xp,
                                int nin, int k, int kp, int d) {
  int i = blockIdx.x, g = blockIdx.y;
  float* dst = Xp + ((size_t)g * kp + i) * d;
  if (i < k) {
    int src = idx[(size_t)g * k + i];
    float v = vals[(size_t)g * k + i];
    const float* s = Hsrc + ((size_t)g * nin + src) * d;
    for (int c = threadIdx.x; c < d; c += blockDim.x) dst[c] = s[c] * v;
  } else {
    for (int c = threadIdx.x; c < d; c += blockDim.x) dst[c] = 0.f;
  }
}

// A1[g,i,j] = A[g, idx[g,i], idx[g,j]] for i,j<k; zero-padded to kp x kp
// grid = (kp, G), block = 256
__global__ void gather_adj_kernel(const float* __restrict__ A, const int* __restrict__ idx,
                                  float* __restrict__ A1, int nin, int k, int kp) {
  int i = blockIdx.x, g = blockIdx.y;
  const int* ig = idx + (size_t)g * k;
  float* dst = A1 + ((size_t)g * kp + i) * kp;
  if (i < k) {
    const float* s = A + ((size_t)g * nin + ig[i]) * nin;
    for (int j = threadIdx.x; j < kp; j += blockDim.x)
      dst[j] = (j < k) ? s[ig[j]] : 0.f;
  } else {
    for (int j = threadIdx.x; j < kp; j += blockDim.x) dst[j] = 0.f;
  }
}

// Pad Wih [512,65]->[512,68] and Wfc [128,7]->[128,16] with zeros.
// grid = 512+128, block = 96
__global__ void pad_weights_kernel(const float* __restrict__ Wih, const float* __restrict__ Wfc,
                                   float* __restrict__ Wihp, float* __restrict__ Wfcp) {
  int bid = blockIdx.x, c = threadIdx.x;
  if (bid < 4 * Hh) {
    if (c < INPAD)
      Wihp[(size_t)bid * INPAD + c] = (c < INLSTM) ? Wih[(size_t)bid * INLSTM + c] : 0.f;
  } else {
    int r = bid - 4 * Hh;
    if (c < 16)
      Wfcp[(size_t)r * 16 + c] = (c < Yy) ? Wfc[(size_t)r * Yy + c] : 0.f;
  }
}

// seq[r,t,c]: r=b*512+nn; c<64 -> pooled row (nn<K2) else 0; c==64 -> X[b,t,nn,63]
// grid = 4096, block = 96 (65 active), loops t
__global__ void build_seq_kernel(const float* __restrict__ H2, const int* __restrict__ idx2,
                                 const float* __restrict__ vals2, const float* __restrict__ X,
                                 float* __restrict__ seq) {
  int r = blockIdx.x;
  int b = r >> 9, nn = r & 511;
  int c = threadIdx.x;
  if (c >= INLSTM) return;
  for (int t = 0; t < Tt; ++t) {
    int g = b * Tt + t;
    float v;
    if (c == OUT) {
      v = X[(((size_t)b * Tt + t) * Nn + nn) * Ff + (Ff - 1)];
    } else if (nn < K2) {
      int src = idx2[(size_t)g * K2 + nn];
      v = H2[((size_t)g * K1 + src) * OUT + c] * vals2[(size_t)g * K2 + nn];
    } else {
      v = 0.f;
    }
    seq[((size_t)r * Tt + t) * INLSTM + c] = v;
  }
}

// ---------------------------------------------------------------------------
// LSTM: 16 sequences per block, 8 waves; gates via WMMA against padded Wih^T /
// Whh^T (each LDS A-pair feeds 4 WMMAs); fused FC (h @ Wfc + bfc) -> d_out.
// grid = 256, block = 256
// ---------------------------------------------------------------------------
__global__ __launch_bounds__(256) void lstm_kernel(
    const float* __restrict__ seq, const float* __restrict__ Wihp,
    const float* __restrict__ Whh, const float* __restrict__ bih,
    const float* __restrict__ bhh, const float* __restrict__ Wfcp,
    const float* __restrict__ bfc, float* __restrict__ out) {
  __shared__ float xt[16][INPAD];   // input, padded 65->68 with zeros
  __shared__ float hb[16][Hh];
  __shared__ float cb[16][Hh];
  __shared__ float gl[16][4 * Hh];  // gates (i,f,g,o)
  int tid = threadIdx.x, blk = blockIdx.x;
  int wave = tid >> 5, lane = tid & 31, half = lane >> 4, l15 = lane & 15;
  int jbase = wave * 64 + l15;      // this wave owns gate columns [wave*64, wave*64+64)

  for (int e = tid; e < 16 * Hh; e += 256) { hb[e >> 7][e & 127] = 0.f; cb[e >> 7][e & 127] = 0.f; }
  __syncthreads();

  for (int t = 0; t < Tt; ++t) {
    for (int e = tid; e < 16 * INPAD; e += 256) {
      int s = e / INPAD, cc = e % INPAD;
      xt[s][cc] = (cc < INLSTM)
          ? seq[(((size_t)(blk * 16 + s)) * Tt + t) * INLSTM + cc] : 0.f;
    }
    __syncthreads();

    v8f acc[4];
#pragma unroll
    for (int q = 0; q < 4; ++q) acc[q] = (v8f){};

#pragma unroll 2
    for (int k0 = 0; k0 < INPAD; k0 += 4) {       // input contraction (K=65 padded)
      int ka = k0 + 2 * half;
      v2f a = *(const v2f*)&xt[l15][ka];
#pragma unroll
      for (int q = 0; q < 4; ++q) {
        v2f b = *(const v2f*)&Wihp[(size_t)(jbase + q * 16) * INPAD + ka];
        acc[q] = wmma4(a, b, acc[q]);
      }
    }
#pragma unroll 2
    for (int k0 = 0; k0 < Hh; k0 += 4) {          // hidden contraction (K=128)
      int ka = k0 + 2 * half;
      v2f a = *(const v2f*)&hb[l15][ka];
#pragma unroll
      for (int q = 0; q < 4; ++q) {
        v2f b = *(const v2f*)&Whh[(size_t)(jbase + q * 16) * Hh + ka];
        acc[q] = wmma4(a, b, acc[q]);
      }
    }
#pragma unroll
    for (int q = 0; q < 4; ++q) {
      int j = jbase + q * 16;
      float bias = bih[j] + bhh[j];
#pragma unroll
      for (int r = 0; r < 8; ++r) gl[r + 8 * half][j] = acc[q][r] + bias;
    }
    __syncthreads();

    for (int e = tid; e < 16 * Hh; e += 256) {    // pointwise cell update
      int s = e >> 7, u = e & 127;
      float gi = gl[s][u], gf = gl[s][Hh + u], gg = gl[s][2 * Hh + u], go = gl[s][3 * Hh + u];
      float cc = sigm(gf) * cb[s][u] + sigm(gi) * tanhf(gg);
      cb[s][u] = cc;
      hb[s][u] = sigm(go) * tanhf(cc);
    }
    __syncthreads();
  }

  if (wave == 0) {                  // fused FC: [16,128] @ [128,16(pad 7)] + bfc
    v8f acc = {};
#pragma unroll 2
    for (int k0 = 0; k0 < Hh; k0 += 4) {
      int ka = k0 + 2 * half;
      v2f a = *(const v2f*)&hb[l15][ka];
      v2f b;
      b.x = Wfcp[(size_t)ka * 16 + l15];
      b.y = Wfcp[(size_t)(ka + 1) * 16 + l15];
      acc = wmma4(a, b, acc);
    }
    if (l15 < Yy) {
      float bb = bfc[l15];
#pragma unroll
      for (int r = 0; r < 8; ++r) {
        int s = r + 8 * half;
        out[((size_t)(blk * 16 + s)) * Yy + l15] = acc[r] + bb;
      }
    }
  }
}

// ---------------------------------------------------------------------------
// Host launcher
// ---------------------------------------------------------------------------
extern "C" void kernel_launch(void* const* d_in, const int* in_sizes, int n_in,
                              void* d_out, int out_size, void* d_ws, size_t ws_size,
                              hipStream_t stream) {
  const float* X   = (const float*)d_in[0];
  const float* A   = (const float*)d_in[1];
  const float* W1  = (const float*)d_in[2];
  const float* b1  = (const float*)d_in[3];
  const float* p1  = (const float*)d_in[4];
  const float* W2  = (const float*)d_in[5];
  const float* b2  = (const float*)d_in[6];
  const float* p2  = (const float*)d_in[7];
  const float* Wih = (const float*)d_in[8];
  const float* Whh = (const float*)d_in[9];
  const float* bih = (const float*)d_in[10];
  const float* bhh = (const float*)d_in[11];
  const float* Wfc = (const float*)d_in[12];
  const float* bfc = (const float*)d_in[13];
  float* out = (float*)d_out;
  float* ws  = (float*)d_ws;

  // workspace layout (floats), slots reused across pipeline stages
  size_t off = 0;
  auto alloc = [&](size_t n) { size_t o = off; off += (n + 255) & ~(size_t)255; return o; };
  size_t slotA = alloc((size_t)Gg * Nn * HID);    // XW1 -> Xp1(pad) -> H2
  size_t slotB = alloc((size_t)Gg * Nn * HID);    // H1  -> XW2(pad)
  size_t slotC = alloc((size_t)Gg * K1P * K1P);   // A1(pad) -> seq
  size_t o_dinv1 = alloc((size_t)Gg * Nn);
  size_t o_dinv2 = alloc((size_t)Gg * K1P);
  size_t o_sc1   = alloc((size_t)Gg * Nn);
  size_t o_sc2   = alloc((size_t)Gg * K1);
  size_t o_v1    = alloc((size_t)Gg * K1);
  size_t o_v2    = alloc((size_t)Gg * K2);
  size_t o_pn    = alloc(16);
  size_t o_wihp  = alloc((size_t)4 * Hh * INPAD); // padded Wih
  size_t o_wfcp  = alloc((size_t)Hh * 16);        // padded Wfc
  size_t o_i1    = alloc((size_t)Gg * K1);        // ints
  size_t o_i2    = alloc((size_t)Gg * K2);        // ints
  int* idx1 = (int*)(ws + o_i1);
  int* idx2 = (int*)(ws + o_i2);

  const int rows1  = Gg * Nn;     // 57344
  const int rows2  = Gg * K1;     // 45920 (compact)
  const int rows2p = Gg * K1P;    // 46592 (padded)

  // 0) pad LSTM/FC weights
  pad_weights_kernel<<<4 * Hh + Hh, 96, 0, stream>>>(Wih, Wfc, ws + o_wihp, ws + o_wfcp);
  // 1) XW1 = X @ W1  [57344,64]@[64,128]
  gemm_wmma_kernel<2><<<dim3(rows1 / 16, HID / 32), 32, 0, stream>>>(
      X, W1, nullptr, ws + slotA, Ff, Ff, HID, HID, 0);
  // 2) ||p1||, ||p2||
  pnorm_kernel<<<2, 128, 0, stream>>>(p1, p2, ws + o_pn);
  // 3) degrees for stage 1
  degree_kernel<<<rows1, 256, 0, stream>>>(A, ws + o_dinv1, Nn);
  // 4) H1 = relu(norm1 @ XW1 + b1)
  adj_gemm_kernel<4><<<dim3(Nn / 16, HID / 64, Gg), 32, 0, stream>>>(
      A, ws + o_dinv1, ws + slotA, b1, ws + slotB, Nn, Nn, HID);
  // 5) score1
  score_kernel<<<rows1 / 8, 256, 0, stream>>>(ws + slotB, p1, ws + o_pn, ws + o_sc1, rows1, HID);
  // 6) top-k 512 -> 410
  topk_kernel<<<Gg, 256, 0, stream>>>(ws + o_sc1, ws + o_v1, idx1, Nn, K1);
  // 7) pooled features Xp1 (padded to 416 rows, slotA)
  gather_x_kernel<<<dim3(K1P, Gg), 128, 0, stream>>>(ws + slotB, idx1, ws + o_v1,
                                                     ws + slotA, Nn, K1, K1P, HID);
  // 8) pooled adjacency A1 (padded 416x416, slotC)
  gather_adj_kernel<<<dim3(K1P, Gg), 256, 0, stream>>>(A, idx1, ws + slotC, Nn, K1, K1P);
  // 9) degrees for stage 2 (over padded A1; pad rows harmless)
  degree_kernel<<<rows2p, 256, 0, stream>>>(ws + slotC, ws + o_dinv2, K1P);
  // 10) XW2 = Xp1 @ W2  [46592,128]@[128,64] (padded rows stay zero, slotB)
  gemm_wmma_kernel<2><<<dim3(rows2p / 16, OUT / 32), 32, 0, stream>>>(
      ws + slotA, W2, nullptr, ws + slotB, HID, HID, OUT, OUT, 0);
  // 11) H2 = relu(norm2 @ XW2 + b2), compact [G,410,64] (slotA)
  adj_gemm_kernel<4><<<dim3(K1P / 16, OUT / 64, Gg), 32, 0, stream>>>(
      ws + slotC, ws + o_dinv2, ws + slotB, b2, ws + slotA, K1P, K1, OUT);
  // 12) score2
  score_kernel<<<rows2 / 8, 256, 0, stream>>>(ws + slotA, p2, ws + o_pn + 1, ws + o_sc2, rows2, OUT);
  // 13) top-k 410 -> 205
  topk_kernel<<<Gg, 256, 0, stream>>>(ws + o_sc2, ws + o_v2, idx2, K1, K2);
  // 14) build LSTM sequences (slotC)
  build_seq_kernel<<<Bb * Nn, 96, 0, stream>>>(ws + slotA, idx2, ws + o_v2, X, ws + slotC);
  // 15) LSTM + FC -> d_out  (raw [4096,7] row-major == reference .view(B,Y,N))
  lstm_kernel<<<(Bb * Nn) / 16, 256, 0, stream>>>(ws + slotC, ws + o_wihp, Whh, bih, bhh,
                                                  ws + o_wfcp, bfc, out);
}